// Wavelet_nD_Recon_75909251989987
// MI455X (gfx1250) — compile-verified
//
#include <hip/hip_runtime.h>

typedef __attribute__((ext_vector_type(2))) float v2f;
typedef __attribute__((ext_vector_type(8))) float v8f;

#define LROW 257  // 256 + 1 pad: column stride becomes conflict-free mod 64 banks

// ---------------------------------------------------------------------------
// Pass 1: inverse-DWT along axis 1 (stride 256*256). One thread per (b,x2,x3)
// column; rolling 4-tap s/w register window, periodic wrap via preload of
// rows 125..127. Fully coalesced along x3.
// ---------------------------------------------------------------------------
__global__ __launch_bounds__(256) void wav_axis1(const float* __restrict__ in,
                                                 const float* __restrict__ hh,
                                                 float* __restrict__ out) {
  const long S1 = 256L * 256L;
  const long base = ((long)blockIdx.y * 256L) * S1 + (long)blockIdx.x * 256L
                  + threadIdx.x;
  const float* p = in + base;
  float* q = out + base;

  const float ts00=hh[6], ts01=hh[4], ts02=hh[2], ts03=hh[0];
  const float ts10=hh[7], ts11=hh[5], ts12=hh[3], ts13=hh[1];
  const float tw00=hh[1], tw01=hh[3], tw02=hh[5], tw03=hh[7];
  const float tw10=-hh[0], tw11=-hh[2], tw12=-hh[4], tw13=-hh[6];

  float s0=p[125*S1], s1=p[126*S1], s2=p[127*S1];
  float w0=p[253*S1], w1=p[254*S1], w2=p[255*S1];
  #pragma unroll 4
  for (int e = 0; e < 128; ++e) {
    const float s3 = p[(long)e * S1];
    const float w3 = p[(long)(e + 128) * S1];
    const float ye = s0*ts00 + s1*ts01 + s2*ts02 + s3*ts03
                   + w0*tw00 + w1*tw01 + w2*tw02 + w3*tw03;
    const float yo = s0*ts10 + s1*ts11 + s2*ts12 + s3*ts13
                   + w0*tw10 + w1*tw11 + w2*tw12 + w3*tw13;
    q[(long)(2*e)     * S1] = ye;
    q[(long)(2*e + 1) * S1] = yo;
    s0=s1; s1=s2; s2=s3;  w0=w1; w1=w2; w2=w3;
  }
}

// ---------------------------------------------------------------------------
// Pass 2: fused axis-2 + axis-3 on a 256x256 slab held in LDS (257-padded).
// Stage B: axis-2 on VALU, each wave owns 32 columns (in-wave read-then-write
// keeps in-place safe). Stage C: axis-3 as banded GEMM on the matrix pipe,
// 6x v_wmma_f32_16x16x4_f32 per 16x16 output tile; D stored straight to HBM.
// ---------------------------------------------------------------------------
__global__ __launch_bounds__(256) void wav_axis23(float* __restrict__ data,
                                                  const float* __restrict__ hh) {
  extern __shared__ float lds[];
  const int tid  = threadIdx.x;
  const int lane = tid & 31;
  const int wave = tid >> 5;
  float* g = data + (long)blockIdx.x * 65536L;

  // ---- Stage A: coalesced global -> LDS --------------------------------
  #pragma unroll 4
  for (int r = 0; r < 256; ++r)
    lds[r * LROW + tid] = g[r * 256 + tid];
  __syncthreads();

  // ---- Stage B: axis-2 transform, in place, wave-owned columns ---------
  {
    const int par = lane & 1;              // output parity fixed per lane
    const float cs0 = par ? hh[7] : hh[6];
    const float cs1 = par ? hh[5] : hh[4];
    const float cs2 = par ? hh[3] : hh[2];
    const float cs3 = par ? hh[1] : hh[0];
    const float cw0 = par ? -hh[0] : hh[1];
    const float cw1 = par ? -hh[2] : hh[3];
    const float cw2 = par ? -hh[4] : hh[5];
    const float cw3 = par ? -hh[6] : hh[7];
    const int ebase = lane >> 1;
    for (int ci = 0; ci < 32; ++ci) {
      const int c = wave * 32 + ci;
      float y[8];
      #pragma unroll
      for (int i = 0; i < 8; ++i) {        // outputs n = lane + 32*i
        const int e  = ebase + 16 * i;
        const int r0 = (e + 125) & 127, r1 = (e + 126) & 127;
        const int r2 = (e + 127) & 127, r3 = e;
        const float sv0 = lds[r0*LROW + c], sv1 = lds[r1*LROW + c];
        const float sv2 = lds[r2*LROW + c], sv3 = lds[r3*LROW + c];
        const float wv0 = lds[(r0+128)*LROW + c], wv1 = lds[(r1+128)*LROW + c];
        const float wv2 = lds[(r2+128)*LROW + c], wv3 = lds[(r3+128)*LROW + c];
        y[i] = sv0*cs0 + sv1*cs1 + sv2*cs2 + sv3*cs3
             + wv0*cw0 + wv1*cw1 + wv2*cw2 + wv3*cw3;
      }
      #pragma unroll
      for (int i = 0; i < 8; ++i)
        lds[(lane + 32*i) * LROW + c] = y[i];
    }
  }
  __syncthreads();

  // ---- Stage C: axis-3 via V_WMMA_F32_16X16X4_F32 ----------------------
  const int laneN = lane & 15;   // A: row M; B/D: column N
  const int half  = lane >> 4;

  // Tile-invariant 12x16 coefficient fragments (B-matrix), built once.
  // B layout: vgpr v, lane half h -> K = 2h+v ; col N = lane%16.
  v2f Bs[3], Bw[3];
  #pragma unroll
  for (int qk = 0; qk < 3; ++qk) {
    #pragma unroll
    for (int v = 0; v < 2; ++v) {
      const int dt = 4*qk + 2*half + v;
      const int j  = dt - (laneN >> 1);
      const int par = laneN & 1;
      float bs = 0.f, bw = 0.f;
      if (j >= 0 && j < 4) {
        bs = hh[(par ? 7 : 6) - 2*j];
        bw = par ? -hh[2*j] : hh[2*j + 1];
      }
      Bs[qk][v] = bs;
      Bw[qk][v] = bw;
    }
  }

  for (int t = 0; t < 2; ++t) {
    const int m0 = (wave * 2 + t) * 16;            // wave-owned row tile
    const float* arow = &lds[(m0 + laneN) * LROW]; // A: lane reads its row
    v8f acc[16];
    #pragma unroll
    for (int n0i = 0; n0i < 16; ++n0i) {           // 16 output col tiles
      const int e0 = 8 * n0i;
      v8f accv = {0.f,0.f,0.f,0.f,0.f,0.f,0.f,0.f};
      #pragma unroll
      for (int qk = 0; qk < 3; ++qk) {
        // A fragment col: t = e0-3+dt (mod 128); dt = 4*qk + 2*half + v
        const int c0 = (e0 + 125 + 4*qk + 2*half) & 127;
        const int c1 = (c0 + 1) & 127;
        v2f a;
        a[0] = arow[c0];       a[1] = arow[c1];        // s half
        accv = __builtin_amdgcn_wmma_f32_16x16x4_f32(
                 false, a, false, Bs[qk], (short)0, accv, false, false);
        a[0] = arow[128 + c0]; a[1] = arow[128 + c1];  // w half
        accv = __builtin_amdgcn_wmma_f32_16x16x4_f32(
                 false, a, false, Bw[qk], (short)0, accv, false, false);
      }
      acc[n0i] = accv;
    }
    // D layout: row = m0 + 8*half + v, col = 16*n0i + laneN
    float* go = g + (long)(m0 + 8*half) * 256 + laneN;
    #pragma unroll
    for (int n0i = 0; n0i < 16; ++n0i) {
      #pragma unroll
      for (int v = 0; v < 8; ++v)
        go[(long)v * 256 + 16*n0i] = acc[n0i][v];
    }
  }
}

// ---------------------------------------------------------------------------
extern "C" void kernel_launch(void* const* d_in, const int* in_sizes, int n_in,
                              void* d_out, int out_size, void* d_ws, size_t ws_size,
                              hipStream_t stream) {
  const float* x = (const float*)d_in[0];
  const float* h = (const float*)d_in[1];
  float* out = (float*)d_out;

  // Pass 1: axis-1, x -> out
  dim3 g1(256, 4), b1(256);
  wav_axis1<<<g1, b1, 0, stream>>>(x, h, out);

  // Pass 2: fused axis-2 + axis-3, in place on out (one 256KB slab per block)
  const size_t shmem = (size_t)256 * LROW * sizeof(float);  // 263168 B < 320KB
  dim3 g2(1024), b2(256);
  wav_axis23<<<g2, b2, shmem, stream>>>(out, h);
}